// TTLinear_14147622273724
// MI455X (gfx1250) — compile-verified
//
#include <hip/hip_runtime.h>

// CDNA5 / gfx1250 fused TT-linear kernel, round 3 (fix async builtin arg types).
//  - fp32 WMMA (V_WMMA_F32_16X16X4_F32) for the three K=128 contractions
//  - double-buffered x-slice staging via GLOBAL_LOAD_ASYNC_TO_LDS (guarded,
//    falls back to plain load+ds_store if the builtin is unavailable)
//  - 2 barriers per inner iteration
//  - fresh WMMA accumulations start from inline-0 C operand (no v_mov burst)
//  - grid = (8 j0 slices, B/16 tiles): j0 fastest-varying for L2 reuse of x

typedef float v2f __attribute__((ext_vector_type(2)));
typedef float v8f __attribute__((ext_vector_type(8)));

#if defined(__has_builtin)
#if __has_builtin(__builtin_amdgcn_global_load_async_to_lds_b32) && \
    __has_builtin(__builtin_amdgcn_s_wait_asynccnt)
#define USE_ASYNC_LDS 1
#endif
#endif
#ifndef USE_ASYNC_LDS
#define USE_ASYNC_LDS 0
#endif

// Builtin signature (from hipcc diagnostic): param 0 is a global-AS int*
// ("__device__ int *" in HIP diagnostic spelling), param 1 the LDS-side int*.
typedef __attribute__((address_space(1))) int gint_t;
typedef __attribute__((address_space(3))) int lint_t;
// Generic->AS casts via integer casts: for LDS the low 32 bits of a generic
// pointer are the hardware LDS byte offset (ISA §10.2 aperture rules).
#define TO_GLOBAL(p) ((gint_t*)(unsigned long long)(p))
#define TO_LDS(p)    ((lint_t*)(unsigned int)(unsigned long long)(p))

// LDS layout (floats):
//   Xs  [2][16][66]          double-buffered x slice        (padded)
//   T0  [16][132]            t0[b][K=s1*8+i1]               (padded, K=128)
//   WS  [8 waves][16][17]    per-wave transpose scratch     (padded)
//   G0s [8][16]              core0 slice at fixed j0
//   G3s [1024]               core3 raw copy
//   G2T [(i2*128+N)*18 + s2] core2 transposed, K-stride 18  (padded)
#define XS_OFF   0      // 2*1056
#define T0_OFF   2112   // 2112
#define WS_OFF   4224   // 2176
#define G0_OFF   6400   // 128
#define G3_OFF   6528   // 1024
#define G2_OFF   7552   // 18432
#define SMEM_FLOATS 25984   // 103936 bytes

static __device__ __forceinline__ v8f vzero8() { v8f z = {}; return z; }

static __device__ __forceinline__ v8f wmma4(v2f a, v2f b, v8f c) {
  // D = A(16x4 f32) * B(4x16 f32) + C(16x16 f32)
  return __builtin_amdgcn_wmma_f32_16x16x4_f32(
      false, a, false, b, (short)0, c, false, false);
}

__global__ __launch_bounds__(256, 1)
void tt_linear_wmma_kernel(const float* __restrict__ x,
                           const float* __restrict__ g0,   // [1,8,8,16]
                           const float* __restrict__ g1,   // [16,8,8,16]
                           const float* __restrict__ g2,   // [16,8,8,16]
                           const float* __restrict__ g3,   // [16,8,8,1]
                           const float* __restrict__ bias, // [1]
                           float* __restrict__ out,
                           int batch) {
  extern __shared__ float sm[];
  float* Xs  = sm + XS_OFF;
  float* T0  = sm + T0_OFF;
  float* WS  = sm + WS_OFF;
  float* G0s = sm + G0_OFF;
  float* G3s = sm + G3_OFF;
  float* G2T = sm + G2_OFF;

  const int tid  = threadIdx.x;
  const int w    = tid >> 5;        // wave id == j1
  const int lane = tid & 31;
  const int half = lane >> 4;
  const int ln   = lane & 15;
  const int j0    = blockIdx.x;     // fastest-varying: 8 j0 blocks share x tile
  const int bBase = blockIdx.y << 4;
  if (bBase + 16 > batch) return;

  // ---- issue x-slice loads for flat iteration m into LDS buffer ----
  auto issueX = [&](int m, float* buf) {
    const int i3 = m >> 3, i2 = m & 7;
#pragma unroll
    for (int q = 0; q < 4; ++q) {
      int e   = tid + q * 256;               // 0..1023
      int b   = e >> 6;
      int pos = e & 63;                      // i0*8 + i1
      int i0 = pos >> 3, i1 = pos & 7;
      const float* gp =
          x + (size_t)(bBase + b) * 4096 + i0 * 512 + i1 * 64 + i2 * 8 + i3;
      float* lp = buf + b * 66 + pos;
#if USE_ASYNC_LDS
      __builtin_amdgcn_global_load_async_to_lds_b32(TO_GLOBAL(gp), TO_LDS(lp),
                                                    0, 0);
#else
      *lp = *gp;
#endif
    }
  };

  // ---- preload LDS-resident operands (once per block) ----
  if (tid < 128) {                          // G0 slice at this j0
    int i0 = tid >> 4, s1 = tid & 15;
    G0s[tid] = g0[i0 * 128 + j0 * 16 + s1];
  }
  for (int e = tid; e < 1024; e += 256)     // G3 raw
    G3s[e] = g3[e];
  for (int e = tid; e < 16384; e += 256) {  // G2 transposed: [i2*128+N][K=s2]
    int s2  = e >> 10;
    int rem = e & 1023;                     // i2*128 + (j2*16+s3)
    G2T[rem * 18 + s2] = g2[e];
  }

  // ---- per-wave register preload of G1 B-fragments (K=128, N-tile j1=w) ----
  v2f g1b[32];
#pragma unroll
  for (int kk = 0; kk < 32; ++kk) {
    int K0 = kk * 4 + half * 2;             // K = s1*8 + i1, N = w*16 + ln
    g1b[kk].x = g1[(K0 >> 3) * 1024 + (K0 & 7) * 128 + w * 16 + ln];
    int K1 = K0 + 1;
    g1b[kk].y = g1[(K1 >> 3) * 1024 + (K1 & 7) * 128 + w * 16 + ln];
  }

  const float bv = bias[0];

  v8f acc2[8];   // t2[b][(j2,s3)] accumulated over i2, j1 = w
  v8f yacc[8];   // y[b][(j2,j3)]  accumulated over (i3,s3), j1 = w
#pragma unroll
  for (int t = 0; t < 8; ++t) { acc2[t] = vzero8(); yacc[t] = vzero8(); }

  issueX(0, Xs);                            // prologue: fill buffer 0
  __syncthreads();                          // also covers G0s/G3s/G2T fills

  float* WSw = WS + w * (16 * 17);

  for (int m = 0; m < 64; ++m) {            // m = i3*8 + i2
    const int i2 = m & 7;
    float* cur = Xs + (m & 1) * 1056;

    if (m < 63) {
      issueX(m + 1, Xs + ((m + 1) & 1) * 1056);   // prefetch next slice
#if USE_ASYNC_LDS
      __builtin_amdgcn_s_wait_asynccnt(4);        // current 4 done, next 4 fly
#endif
    } else {
#if USE_ASYNC_LDS
      __builtin_amdgcn_s_wait_asynccnt(0);
#endif
    }
    __syncthreads();   // barrier A: cur complete; prev iter's T0 readers done

    // ---- step0 (VALU, K=8): T0[b][s1*8+i1] = sum_i0 x * G0 ----
    {
      int b = tid >> 4, sub = tid & 15;
#pragma unroll
      for (int q = 0; q < 8; ++q) {
        int K  = sub + q * 16;
        int s1 = K >> 3, i1 = K & 7;
        float acc = 0.f;
#pragma unroll
        for (int i0 = 0; i0 < 8; ++i0)
          acc += cur[b * 66 + i0 * 8 + i1] * G0s[i0 * 16 + s1];
        T0[b * 132 + K] = acc;
      }
    }
    __syncthreads();   // barrier B: T0 ready for all waves

    // ---- step1 (WMMA, M16 N16 K128): t1[b][s2] at j1=w ----
    v8f c1 = vzero8();
#pragma unroll
    for (int kk = 0; kk < 32; ++kk) {
      int K0 = kk * 4 + half * 2;
      v2f a;
      a.x = T0[ln * 132 + K0];
      a.y = T0[ln * 132 + K0 + 1];
      c1 = wmma4(a, g1b[kk], c1);
    }

    // ---- D-layout -> A-layout via per-wave scratch (in-order LDS) ----
#pragma unroll
    for (int r = 0; r < 8; ++r)
      WSw[(half * 8 + r) * 17 + ln] = c1[r];
    v2f a2[4];
#pragma unroll
    for (int kk = 0; kk < 4; ++kk) {
      int K0 = kk * 4 + half * 2;
      a2[kk].x = WSw[ln * 17 + K0];
      a2[kk].y = WSw[ln * 17 + K0 + 1];
    }

    // ---- step2 (WMMA, K=16 per i2, accumulate over i2 in regs) ----
    if (i2 == 0) {                       // fresh accumulation: inline-0 C
#pragma unroll
      for (int j2 = 0; j2 < 8; ++j2) {
#pragma unroll
        for (int kk = 0; kk < 4; ++kk) {
          int K0 = kk * 4 + half * 2;    // = s2
          int N  = j2 * 16 + ln;         // (j2, s3)
          v2f bfr;
          bfr.x = G2T[(i2 * 128 + N) * 18 + K0];
          bfr.y = G2T[(i2 * 128 + N) * 18 + K0 + 1];
          acc2[j2] = wmma4(a2[kk], bfr, kk == 0 ? vzero8() : acc2[j2]);
        }
      }
    } else {
#pragma unroll
      for (int j2 = 0; j2 < 8; ++j2) {
#pragma unroll
        for (int kk = 0; kk < 4; ++kk) {
          int K0 = kk * 4 + half * 2;
          int N  = j2 * 16 + ln;
          v2f bfr;
          bfr.x = G2T[(i2 * 128 + N) * 18 + K0];
          bfr.y = G2T[(i2 * 128 + N) * 18 + K0 + 1];
          acc2[j2] = wmma4(a2[kk], bfr, acc2[j2]);
        }
      }
    }

    // ---- step3 (WMMA, K=16 = s3): fires after the last i2 of each i3 ----
    if (i2 == 7) {
      const int i3 = m >> 3;
      v2f b3[4];
#pragma unroll
      for (int kk = 0; kk < 4; ++kk) {
        int K0 = kk * 4 + half * 2;      // = s3
        b3[kk].x = (ln < 8) ? G3s[K0 * 64 + i3 * 8 + ln] : 0.f;
        b3[kk].y = (ln < 8) ? G3s[(K0 + 1) * 64 + i3 * 8 + ln] : 0.f;
      }
#pragma unroll
      for (int j2 = 0; j2 < 8; ++j2) {
#pragma unroll
        for (int r = 0; r < 8; ++r)      // transpose t2 tile via scratch
          WSw[(half * 8 + r) * 17 + ln] = acc2[j2][r];
#pragma unroll
        for (int kk = 0; kk < 4; ++kk) {
          int K0 = kk * 4 + half * 2;
          v2f a3;
          a3.x = WSw[ln * 17 + K0];
          a3.y = WSw[ln * 17 + K0 + 1];
          yacc[j2] = wmma4(a3, b3[kk], yacc[j2]);
        }
        // acc2[j2] restarts from inline-0 C at the next i2==0; no re-zeroing
      }
    }
  }

  // ---- epilogue: bias + relu, store ----
  if (ln < 8) {
#pragma unroll
    for (int j2 = 0; j2 < 8; ++j2) {
#pragma unroll
      for (int r = 0; r < 8; ++r) {
        int b = half * 8 + r;
        float v = yacc[j2][r] + bv;
        v = v > 0.f ? v : 0.f;
        out[(size_t)(bBase + b) * 4096 + j0 * 512 + w * 64 + j2 * 8 + ln] = v;
      }
    }
  }
}

extern "C" void kernel_launch(void* const* d_in, const int* in_sizes, int n_in,
                              void* d_out, int out_size, void* d_ws, size_t ws_size,
                              hipStream_t stream) {
  const float* x  = (const float*)d_in[0];
  const float* g0 = (const float*)d_in[1];
  const float* g1 = (const float*)d_in[2];
  const float* g2 = (const float*)d_in[3];
  const float* g3 = (const float*)d_in[4];
  const float* bs = (const float*)d_in[5];
  float* out = (float*)d_out;

  int B = in_sizes[0] / 4096;                  // 16384
  dim3 grid(8, B / 16, 1);                     // j0 fastest for L2 x reuse
  dim3 block(256, 1, 1);                       // 8 wave32s
  size_t smem = SMEM_FLOATS * sizeof(float);   // ~101.5 KB dynamic LDS
  tt_linear_wmma_kernel<<<grid, block, smem, stream>>>(x, g0, g1, g2, g3, bs,
                                                       out, B);
}